// AttentionOT_71399536328891
// MI455X (gfx1250) — compile-verified
//
#include <hip/hip_runtime.h>
#include <math.h>

// ---------------- problem constants (fixed by reference) ----------------
#define BB    8
#define NQv   1024
#define NKv   1024
#define DIMv  512

#define EPS_SINK 0.05f
#define INV_EPS  20.0f
#define THRESHv  1e-3f
#define MAX_ITERv 100

typedef __attribute__((ext_vector_type(2))) float v2f;
typedef __attribute__((ext_vector_type(4))) float v4f;
typedef __attribute__((ext_vector_type(8))) float v8f;

// workspace layout (in floats)
static const long OFF_XQ   = 0;
static const long OFF_XK   = OFF_XQ  + (long)BB * NQv * DIMv;   //  4,194,304
static const long OFF_VP   = OFF_XK  + (long)BB * NKv * DIMv;   //  8,388,608
static const long OFF_SIM  = OFF_VP  + (long)BB * NKv * DIMv;   // 12,582,912
static const long OFF_SIMT = OFF_SIM + (long)BB * NKv * NQv;    // 20,971,520
static const long OFF_U    = OFF_SIMT+ (long)BB * NQv * NKv;    // 29,360,128
static const long OFF_V    = OFF_U   + (long)BB * NKv;
static const long OFF_ERR  = OFF_V   + (long)BB * NQv;
static const long OFF_DONE = OFF_ERR + 1;

// ---------------- fp32 WMMA helper (CDNA5 V_WMMA_F32_16X16X4_F32) -------
__device__ __forceinline__ v8f wmma_f32(v2f a, v2f b, v8f c) {
  // 8 args: (neg_a, A, neg_b, B, c_mod, C, reuse_a, reuse_b)
  return __builtin_amdgcn_wmma_f32_16x16x4_f32(
      false, a, false, b, (short)0, c, false, false);
}

// ---------------- generic C = A(MxK) * Bt(NxK)^T (+bias) ----------------
// grid: (Ndim/64, Mdim/128, BB), block 256. Bt is B^T stored row-major [N,K].
__global__ __launch_bounds__(256) void gemm_abt(
    const float* __restrict__ A, const float* __restrict__ Bt,
    float* __restrict__ C, const float* __restrict__ bias,
    int Ndim, int Kdim, long sA, long sBt, long sC)
{
  __shared__ __align__(16) float As[128][20];
  __shared__ __align__(16) float Bs[64][20];
  const int bz = blockIdx.z;
  A  += (long)bz * sA;
  Bt += (long)bz * sBt;
  C  += (long)bz * sC;

  const int t    = threadIdx.x;
  const int lane = t & 31;
  const int wave = t >> 5;
  const int wm = (wave & 3) * 32;           // 4 waves along M
  const int wn = (wave >> 2) * 32;          // 2 waves along N
  const int blockM = blockIdx.y * 128;
  const int blockN = blockIdx.x * 64;
  const int row16 = lane & 15;
  const int kSel  = (lane >> 4) << 1;       // 0 or 2
  const int cRow  = (lane >> 4) << 3;       // 0 or 8

  v8f zero = {};
  v8f acc[2][2];
#pragma unroll
  for (int i = 0; i < 2; ++i)
#pragma unroll
    for (int j = 0; j < 2; ++j) acc[i][j] = zero;

  for (int k0 = 0; k0 < Kdim; k0 += 16) {
    // A tile 128x16 : 512 float4, 2 per thread
#pragma unroll
    for (int i = 0; i < 2; ++i) {
      int s = t + i * 256;
      int r = s >> 2, c4 = (s & 3) << 2;
      *(v4f*)&As[r][c4] = *(const v4f*)&A[(long)(blockM + r) * Kdim + k0 + c4];
    }
    // Bt tile 64x16 : 256 float4, 1 per thread
    {
      int r = t >> 2, c4 = (t & 3) << 2;
      *(v4f*)&Bs[r][c4] = *(const v4f*)&Bt[(long)(blockN + r) * Kdim + k0 + c4];
    }
    __syncthreads();
#pragma unroll
    for (int kk = 0; kk < 16; kk += 4) {
      v2f af0 = *(const v2f*)&As[wm      + row16][kk + kSel];
      v2f af1 = *(const v2f*)&As[wm + 16 + row16][kk + kSel];
      v2f bf0 = *(const v2f*)&Bs[wn      + row16][kk + kSel];
      v2f bf1 = *(const v2f*)&Bs[wn + 16 + row16][kk + kSel];
      acc[0][0] = wmma_f32(af0, bf0, acc[0][0]);
      acc[0][1] = wmma_f32(af0, bf1, acc[0][1]);
      acc[1][0] = wmma_f32(af1, bf0, acc[1][0]);
      acc[1][1] = wmma_f32(af1, bf1, acc[1][1]);
    }
    __syncthreads();
  }

#pragma unroll
  for (int i = 0; i < 2; ++i)
#pragma unroll
    for (int j = 0; j < 2; ++j) {
      int gn = blockN + wn + j * 16 + row16;
      float bv = bias ? bias[gn] : 0.0f;
#pragma unroll
      for (int r = 0; r < 8; ++r) {
        int gm = blockM + wm + i * 16 + cRow + r;
        C[(long)gm * Ndim + gn] = acc[i][j][r] + bv;
      }
    }
}

// ---- X1 = TT * VP where TT[n][m] = exp((SIMT[n,m]-1+u[m]+v[n])/eps) ----
// grid: (DIMv/64, NQv/128, BB), block 256. VP row-major [NK, DIM] (transposed
// into LDS on store).
__global__ __launch_bounds__(256) void gemm_ttv(
    const float* __restrict__ SIMT, const float* __restrict__ VP,
    const float* __restrict__ U, const float* __restrict__ V,
    float* __restrict__ X1)
{
  __shared__ __align__(16) float As[128][20];
  __shared__ __align__(16) float Bs[64][20];
  const int bz = blockIdx.z;
  const float* St = SIMT + (long)bz * NQv * NKv;
  const float* Vp = VP   + (long)bz * NKv * DIMv;
  const float* Ub = U + bz * NKv;
  const float* Vb = V + bz * NQv;
  float* C = X1 + (long)bz * NQv * DIMv;

  const int t    = threadIdx.x;
  const int lane = t & 31;
  const int wave = t >> 5;
  const int wm = (wave & 3) * 32;
  const int wn = (wave >> 2) * 32;
  const int blockM = blockIdx.y * 128;
  const int blockN = blockIdx.x * 64;
  const int row16 = lane & 15;
  const int kSel  = (lane >> 4) << 1;
  const int cRow  = (lane >> 4) << 3;

  v8f zero = {};
  v8f acc[2][2];
#pragma unroll
  for (int i = 0; i < 2; ++i)
#pragma unroll
    for (int j = 0; j < 2; ++j) acc[i][j] = zero;

  for (int k0 = 0; k0 < NKv; k0 += 16) {
    // A tile from SIMT with fused exp transform
#pragma unroll
    for (int i = 0; i < 2; ++i) {
      int s = t + i * 256;
      int r = s >> 2, c4 = (s & 3) << 2;
      v4f sv = *(const v4f*)&St[(long)(blockM + r) * NKv + k0 + c4];
      float vr = Vb[blockM + r];
#pragma unroll
      for (int j = 0; j < 4; ++j)
        As[r][c4 + j] = __expf((sv[j] - 1.0f + Ub[k0 + c4 + j] + vr) * INV_EPS);
    }
    // B tile from VP (row-major KxN), transpose into Bs[n][k]
    {
      int kr = t >> 4, n4 = (t & 15) << 2;
      v4f bv = *(const v4f*)&Vp[(long)(k0 + kr) * DIMv + blockN + n4];
#pragma unroll
      for (int j = 0; j < 4; ++j) Bs[n4 + j][kr] = bv[j];
    }
    __syncthreads();
#pragma unroll
    for (int kk = 0; kk < 16; kk += 4) {
      v2f af0 = *(const v2f*)&As[wm      + row16][kk + kSel];
      v2f af1 = *(const v2f*)&As[wm + 16 + row16][kk + kSel];
      v2f bf0 = *(const v2f*)&Bs[wn      + row16][kk + kSel];
      v2f bf1 = *(const v2f*)&Bs[wn + 16 + row16][kk + kSel];
      acc[0][0] = wmma_f32(af0, bf0, acc[0][0]);
      acc[0][1] = wmma_f32(af0, bf1, acc[0][1]);
      acc[1][0] = wmma_f32(af1, bf0, acc[1][0]);
      acc[1][1] = wmma_f32(af1, bf1, acc[1][1]);
    }
    __syncthreads();
  }

#pragma unroll
  for (int i = 0; i < 2; ++i)
#pragma unroll
    for (int j = 0; j < 2; ++j) {
      int gn = blockN + wn + j * 16 + row16;
#pragma unroll
      for (int r = 0; r < 8; ++r) {
        int gm = blockM + wm + i * 16 + cRow + r;
        C[(long)gm * DIMv + gn] = acc[i][j][r];
      }
    }
}

// ---------------- row L2-normalize (one wave32 per row of DIM) ----------
__global__ __launch_bounds__(256) void l2norm_rows(float* __restrict__ X)
{
  int gw   = (blockIdx.x * 256 + threadIdx.x) >> 5;
  int lane = threadIdx.x & 31;
  float* row = X + (long)gw * DIMv;
  float s = 0.0f;
  for (int i = lane; i < DIMv; i += 32) { float x = row[i]; s += x * x; }
#pragma unroll
  for (int off = 16; off > 0; off >>= 1) s += __shfl_xor(s, off, 32);
  float inv = 1.0f / fmaxf(sqrtf(s), 1e-12f);
  for (int i = lane; i < DIMv; i += 32) row[i] *= inv;
}

// ---------------- Sinkhorn u-update: one wave per (b,m) row of SIM ------
__global__ __launch_bounds__(256) void sinkhorn_u(
    const float* __restrict__ SIM, float* __restrict__ U,
    const float* __restrict__ V, const int* __restrict__ done,
    float* __restrict__ errSum)
{
  if (*done) return;
  int gw   = (blockIdx.x * 256 + threadIdx.x) >> 5;   // b*NK + m
  int lane = threadIdx.x & 31;
  int b = gw >> 10;
  const float* row = SIM + (long)gw * NQv;
  const float* vb  = V + b * NQv;
  float um = U[gw];
  float mx = -INFINITY, sm = 0.0f;
  for (int n = lane; n < NQv; n += 32) {
    float tt = (row[n] - 1.0f + um + vb[n]) * INV_EPS;
    float m2 = fmaxf(mx, tt);
    sm = sm * __expf(mx - m2) + __expf(tt - m2);
    mx = m2;
  }
#pragma unroll
  for (int off = 16; off > 0; off >>= 1) {
    float mo = __shfl_xor(mx, off, 32);
    float so = __shfl_xor(sm, off, 32);
    float m2 = fmaxf(mx, mo);
    sm = sm * __expf(mx - m2) + so * __expf(mo - m2);
    mx = m2;
  }
  float lse   = mx + __logf(sm);
  float logMu = __logf(1.0f / (float)NKv + 1e-8f);
  float uNew  = EPS_SINK * (logMu - lse) + um;
  if (lane == 0) {
    U[gw] = uNew;
    atomicAdd(errSum, fabsf(uNew - um));
  }
}

// ---------------- Sinkhorn v-update: one wave per (b,n) row of SIMT -----
__global__ __launch_bounds__(256) void sinkhorn_v(
    const float* __restrict__ SIMT, const float* __restrict__ U,
    float* __restrict__ V, const int* __restrict__ done)
{
  if (*done) return;
  int gw   = (blockIdx.x * 256 + threadIdx.x) >> 5;   // b*NQ + n
  int lane = threadIdx.x & 31;
  int b = gw >> 10;
  const float* row = SIMT + (long)gw * NKv;
  const float* ub  = U + b * NKv;
  float vn = V[gw];
  float mx = -INFINITY, sm = 0.0f;
  for (int m = lane; m < NKv; m += 32) {
    float tt = (row[m] - 1.0f + ub[m] + vn) * INV_EPS;
    float m2 = fmaxf(mx, tt);
    sm = sm * __expf(mx - m2) + __expf(tt - m2);
    mx = m2;
  }
#pragma unroll
  for (int off = 16; off > 0; off >>= 1) {
    float mo = __shfl_xor(mx, off, 32);
    float so = __shfl_xor(sm, off, 32);
    float m2 = fmaxf(mx, mo);
    sm = sm * __expf(mx - m2) + so * __expf(mo - m2);
    mx = m2;
  }
  float lse   = mx + __logf(sm);
  float logNu = __logf(1.0f / (float)NQv + 1e-8f);
  if (lane == 0) V[gw] = EPS_SINK * (logNu - lse) + vn;
}

__global__ void sinkhorn_check(int* done, float* errSum)
{
  if (!*done) {
    if (*errSum * (1.0f / (float)(BB * NKv)) < THRESHv) *done = 1;
  }
  *errSum = 0.0f;
}

// --- attn_save[b,j,k] = 2048 * sum_{n: n%2==j} sim[b,k,n]*T[b,k,n] ------
__global__ __launch_bounds__(256) void attn_reduce(
    const float* __restrict__ SIM, const float* __restrict__ U,
    const float* __restrict__ V, float* __restrict__ attn)
{
  int gw   = (blockIdx.x * 256 + threadIdx.x) >> 5;   // b*NK + k
  int lane = threadIdx.x & 31;
  int b = gw >> 10, k = gw & 1023;
  const float* row = SIM + (long)gw * NQv;
  const float* vb  = V + b * NQv;
  float uk = U[gw];
  float s = 0.0f;
  for (int n = lane; n < NQv; n += 32) {       // n%2 == lane%2 (stride 32)
    float sim = row[n];
    float T = __expf((sim - 1.0f + uk + vb[n]) * INV_EPS);
    s += sim * T;
  }
  // reduce within parity classes (keep bit0 of lane id)
#pragma unroll
  for (int off = 16; off >= 2; off >>= 1) s += __shfl_xor(s, off, 32);
  if (lane < 2)
    attn[(long)b * 2 * NKv + lane * NKv + k] = s * 2048.0f;  // NQ*NK/(NQ/2)
}

// ------------------------------ launcher --------------------------------
extern "C" void kernel_launch(void* const* d_in, const int* in_sizes, int n_in,
                              void* d_out, int out_size, void* d_ws, size_t ws_size,
                              hipStream_t stream)
{
  const float* Q  = (const float*)d_in[0];
  const float* K  = (const float*)d_in[1];
  const float* Vl = (const float*)d_in[2];
  const float* Wq = (const float*)d_in[3];
  const float* Wk = (const float*)d_in[4];
  const float* Wv = (const float*)d_in[5];
  const float* Wp = (const float*)d_in[6];
  const float* bp = (const float*)d_in[7];
  float* out = (float*)d_out;
  float* ws  = (float*)d_ws;

  float* XQ   = ws + OFF_XQ;
  float* XK   = ws + OFF_XK;
  float* VP   = ws + OFF_VP;
  float* SIM  = ws + OFF_SIM;
  float* SIMT = ws + OFF_SIMT;
  float* U    = ws + OFF_U;
  float* V    = ws + OFF_V;
  float* ERR  = ws + OFF_ERR;
  int*   DONE = (int*)(ws + OFF_DONE);
  float* X1   = XQ;  // XQ dead after SIM/SIMT are built -> reuse

  // zero u, v, err, done (contiguous)
  hipMemsetAsync(U, 0, (size_t)(BB * NKv + BB * NQv + 2) * sizeof(float), stream);

  dim3 blk(256);
  const long sQD = (long)NQv * DIMv;   // also NK*DIM (NQ==NK)
  const long sMM = (long)NKv * NQv;

  // projections: X = in @ W^T   (Bt = W, row-major [out,in] == B^T)
  gemm_abt<<<dim3(DIMv/64, NQv/128, BB), blk, 0, stream>>>(Q,  Wq, XQ, nullptr, DIMv, DIMv, sQD, 0L, sQD);
  gemm_abt<<<dim3(DIMv/64, NKv/128, BB), blk, 0, stream>>>(K,  Wk, XK, nullptr, DIMv, DIMv, sQD, 0L, sQD);
  gemm_abt<<<dim3(DIMv/64, NKv/128, BB), blk, 0, stream>>>(Vl, Wv, VP, nullptr, DIMv, DIMv, sQD, 0L, sQD);

  l2norm_rows<<<BB*NQv/8, blk, 0, stream>>>(XQ);
  l2norm_rows<<<BB*NKv/8, blk, 0, stream>>>(XK);

  // sim[b,m,n] = XK . XQ^T ; simT[b,n,m] = XQ . XK^T
  gemm_abt<<<dim3(NQv/64, NKv/128, BB), blk, 0, stream>>>(XK, XQ, SIM,  nullptr, NQv, DIMv, sQD, sQD, sMM);
  gemm_abt<<<dim3(NKv/64, NQv/128, BB), blk, 0, stream>>>(XQ, XK, SIMT, nullptr, NKv, DIMv, sQD, sQD, sMM);

  for (int it = 0; it < MAX_ITERv; ++it) {
    sinkhorn_u<<<BB*NKv/8, blk, 0, stream>>>(SIM,  U, V, DONE, ERR);
    sinkhorn_v<<<BB*NQv/8, blk, 0, stream>>>(SIMT, U, V, DONE);
    sinkhorn_check<<<1, 1, 0, stream>>>(DONE, ERR);
  }

  // attn_save -> second output region
  attn_reduce<<<BB*NKv/8, blk, 0, stream>>>(SIM, U, V, out + (long)BB * NQv * DIMv);

  // X1 = T^T @ VP  (T fused from SIMT,u,v), then out = X1 @ Wp^T + bp
  gemm_ttv<<<dim3(DIMv/64, NQv/128, BB), blk, 0, stream>>>(SIMT, VP, U, V, X1);
  gemm_abt<<<dim3(DIMv/64, NQv/128, BB), blk, 0, stream>>>(X1, Wp, out, bp, DIMv, DIMv, sQD, 0L, sQD);
}